// FieldRewritingProtocol_56650618634581
// MI455X (gfx1250) — compile-verified
//
#include <hip/hip_runtime.h>
#include <math.h>

// Problem dims (fixed by the reference).
#define HH 1024
#define WW 1024
#define NA 64
#define NBLK (64 * 64)   // 16x16 tiles over 1024x1024

typedef __attribute__((ext_vector_type(2))) float v2f;
typedef __attribute__((ext_vector_type(8))) float v8f;

// One ds_swizzle_b32 (group-of-32 xor mode) + add per butterfly step.
// offset encoding: [14:10]=xor_mask, [9:5]=or_mask(0), [4:0]=and_mask(0x1F).
// All 32 lanes are active at every call site, so no bounds clamping needed
// (unlike __shfl_xor, which emits cmp/cndmask/bpermute chains).
#define SWZ_XOR_ADD(r, mask)                                                  \
  (r) += __int_as_float(__builtin_amdgcn_ds_swizzle(                          \
      __float_as_int(r), (((mask) << 10) | 0x1F)))

// ---------------------------------------------------------------------------
// 32-lane wave sum routed through the matrix pipe.
// A = ones(16x4), B holds one f32 per lane in its first VGPR (b.y = 0).
// D = A*B  => every row of D equals the column sums of B, so after the WMMA
// every lane's c[0] == colsum[lane & 15]. Four xor-swizzles over the 16
// columns then give the full 32-lane sum in every lane. Correct regardless
// of the exact K-row striping of B, because we reduce all 16 columns.
// EXEC is all 1s at every call site (full 256-thread blocks, unconditional
// calls), satisfying the WMMA restriction.
// ---------------------------------------------------------------------------
__device__ __forceinline__ float wave_sum32(float x) {
  v2f a; a.x = 1.0f; a.y = 1.0f;
  v2f b; b.x = x;    b.y = 0.0f;
  v8f c = {};
  c = __builtin_amdgcn_wmma_f32_16x16x4_f32(
      /*neg_a=*/false, a, /*neg_b=*/false, b,
      /*c_mod=*/(short)0, c, /*reuse_a=*/false, /*reuse_b=*/false);
  float r = c[0];
  SWZ_XOR_ADD(r, 1);
  SWZ_XOR_ADD(r, 2);
  SWZ_XOR_ADD(r, 4);
  SWZ_XOR_ADD(r, 8);
  return r;
}

// Fast sigmoid: v_exp_f32 + v_rcp_f32 (~1 ulp), avoids the IEEE
// v_div_fmas/v_div_fixup sequence on the per-pixel critical path.
__device__ __forceinline__ float fast_sigmoid(float x) {
  return __builtin_amdgcn_rcpf(1.0f + __expf(-x));
}

// ---------------------------------------------------------------------------
// Pass 1: fused influence map + blend + partial norm reduction.
// One 16x16 pixel tile per 256-thread block. Attractors are pruned per tile
// against the tile bbox (radius sentinel -1 disables), so the inner k-loop
// is ~2 instructions for inactive attractors and summation order is fixed
// (k ascending) -> deterministic.
// ---------------------------------------------------------------------------
__global__ __launch_bounds__(256)
void frp_main(const float* __restrict__ field, const float* __restrict__ sig,
              const int*   __restrict__ apos,  const float* __restrict__ astr,
              const float* __restrict__ p_infl, const float* __restrict__ p_gr,
              const float* __restrict__ p_lr,
              float* __restrict__ out, float* __restrict__ partials) {
  __shared__ int   sAi[NA], sAj[NA];
  __shared__ float sRad[NA], sCoef[NA], sNeg[NA];
  __shared__ float sWave[16];

  const int tx  = threadIdx.x, ty = threadIdx.y;
  const int tid = ty * 16 + tx;
  const int r0  = blockIdx.y * 16;
  const int c0  = blockIdx.x * 16;

  if (tid < NA) {
    const int   ai = apos[2 * tid + 0];
    const int   aj = apos[2 * tid + 1];
    const float s  = astr[tid];
    const float r  = floorf(5.0f * s);          // radius (<= 14)
    const int   ri = (int)r;
    const bool  ov = (ai + ri >= r0) && (ai - ri <= r0 + 15) &&
                     (aj + ri >= c0) && (aj - ri <= c0 + 15);
    sAi[tid]   = ai;
    sAj[tid]   = aj;
    sRad[tid]  = ov ? r : -1.0f;                // -1 => mask always fails
    sCoef[tid] = p_lr[0] * s;                   // local_ratio * s
    sNeg[tid]  = -0.125f / (s * s);             // -0.5 / (4 s^2)
  }
  __syncthreads();

  const int i   = r0 + ty;
  const int j   = c0 + tx;
  const int idx = i * WW + j;

  float contrib = 0.0f;
#pragma unroll 4
  for (int k = 0; k < NA; ++k) {
    const float r   = sRad[k];
    const float adi = fabsf((float)(i - sAi[k]));
    const float adj = fabsf((float)(j - sAj[k]));
    if (adi <= r && adj <= r) {
      const float d2 = adi * adi + adj * adj;
      contrib += sCoef[k] * __expf(d2 * sNeg[k]);
    }
  }

  // sigmoid(gr + contrib), already in (0,1) so the clip is a no-op.
  const float im  = fast_sigmoid(p_gr[0] + contrib);
  const float aa  = im * p_infl[0];
  const float f   = field[idx];
  const float g   = sig[idx];
  const float upd = f * (1.0f - aa) + g * aa;
  out[idx] = upd;

  // Norm partials: WMMA wave reduction, then fixed-order LDS combine.
  const float f2 = wave_sum32(f * f);
  const float u2 = wave_sum32(upd * upd);

  const int wave = tid >> 5;
  const int lane = tid & 31;
  if (lane == 0) { sWave[wave] = f2; sWave[8 + wave] = u2; }
  __syncthreads();
  if (tid == 0) {
    float F = 0.0f, U = 0.0f;
    for (int w = 0; w < 8; ++w) { F += sWave[w]; U += sWave[8 + w]; }
    const int b = blockIdx.y * gridDim.x + blockIdx.x;
    partials[2 * b + 0] = F;
    partials[2 * b + 1] = U;
  }
}

// ---------------------------------------------------------------------------
// Pass 2: reduce the 4096 block partials (fixed strided order -> det.) and
// produce the rescale factor  ||field|| / ||upd||  (1.0 when ||upd|| == 0).
// ---------------------------------------------------------------------------
__global__ __launch_bounds__(256)
void frp_reduce(const float* __restrict__ partials, float* __restrict__ scale) {
  __shared__ float sWave[16];
  const int tid = threadIdx.x;
  float F = 0.0f, U = 0.0f;
  for (int p = tid; p < NBLK; p += 256) {
    F += partials[2 * p + 0];
    U += partials[2 * p + 1];
  }
  F = wave_sum32(F);
  U = wave_sum32(U);
  const int wave = tid >> 5;
  const int lane = tid & 31;
  if (lane == 0) { sWave[wave] = F; sWave[8 + wave] = U; }
  __syncthreads();
  if (tid == 0) {
    float Ft = 0.0f, Ut = 0.0f;
    for (int w = 0; w < 8; ++w) { Ft += sWave[w]; Ut += sWave[8 + w]; }
    const float n = sqrtf(Ut);
    scale[0] = (n > 0.0f) ? (sqrtf(Ft) / n) : 1.0f;
  }
}

// ---------------------------------------------------------------------------
// Pass 3: rescale the output (float4 streaming, bandwidth bound).
// ---------------------------------------------------------------------------
__global__ __launch_bounds__(256)
void frp_scale(float* __restrict__ out, const float* __restrict__ scale) {
  const float s = scale[0];
  const int idx = (blockIdx.x * 256 + threadIdx.x) * 4;
  float4 v = *(float4*)(out + idx);
  v.x *= s; v.y *= s; v.z *= s; v.w *= s;
  *(float4*)(out + idx) = v;
}

extern "C" void kernel_launch(void* const* d_in, const int* in_sizes, int n_in,
                              void* d_out, int out_size, void* d_ws, size_t ws_size,
                              hipStream_t stream) {
  const float* field = (const float*)d_in[0];
  const float* sig   = (const float*)d_in[1];
  const int*   apos  = (const int*)d_in[2];
  const float* astr  = (const float*)d_in[3];
  const float* infl  = (const float*)d_in[4];
  const float* gr    = (const float*)d_in[5];
  const float* lr    = (const float*)d_in[6];
  float* out = (float*)d_out;

  float* partials = (float*)d_ws;          // NBLK * 2 floats (32 KB)
  float* scale    = partials + NBLK * 2;   // 1 float

  dim3 grid(64, 64), block(16, 16);
  frp_main<<<grid, block, 0, stream>>>(field, sig, apos, astr, infl, gr, lr,
                                       out, partials);
  frp_reduce<<<1, 256, 0, stream>>>(partials, scale);
  frp_scale<<<(HH * WW) / (256 * 4), 256, 0, stream>>>(out, scale);
}